// MultiLevel3_33775622815760
// MI455X (gfx1250) — compile-verified
//
#include <hip/hip_runtime.h>
#include <hip/hip_bf16.h>

// ---------------------------------------------------------------------------
// Problem constants (from reference)
// ---------------------------------------------------------------------------
#define N0 50000
#define N1 12500
#define N2 3125
#define E0 800000
#define E1 200000
#define E2 50000
#define NNZ 200000
#define H 128
#define F3 96      // 3*F
#define DECW 64

typedef __attribute__((ext_vector_type(2))) float v2f;
typedef __attribute__((ext_vector_type(8))) float v8f;

// ---------------------------------------------------------------------------
// Utility kernels
// ---------------------------------------------------------------------------
__global__ void zero_kernel(float* __restrict__ p, long long count) {
    long long t = (long long)blockIdx.x * blockDim.x + threadIdx.x;
    if (t < count) p[t] = 0.0f;
}

// conv1: h1[r][c] = relu( sum_k agg6[r][k]*Wrel[k][c] + brel[c] + sum_k x[r][k]*Wroot[k][c] )
// blockDim = 128 (one row per block), K = 6
__global__ void conv1_kernel(const float* __restrict__ x, const float* __restrict__ agg6,
                             const float* __restrict__ Wrel, const float* __restrict__ brel,
                             const float* __restrict__ Wroot, float* __restrict__ h1, int n) {
    int r = blockIdx.x;
    int c = threadIdx.x;
    if (r >= n) return;
    float s = brel[c];
#pragma unroll
    for (int k = 0; k < 6; ++k) {
        s += agg6[r * 6 + k] * Wrel[k * H + c];
        s += x[r * 6 + k] * Wroot[k * H + c];
    }
    h1[r * H + c] = fmaxf(s, 0.0f);
}

// scatter of 6-wide node features over edges: agg6[dst] += x[src]
__global__ void scatter6_kernel(const float* __restrict__ x, const int* __restrict__ ei,
                                int E, float* __restrict__ agg6) {
    int e = blockIdx.x * blockDim.x + threadIdx.x;
    if (e >= E) return;
    int src = ei[e];
    int dst = ei[E + e];
#pragma unroll
    for (int k = 0; k < 6; ++k)
        atomicAdd(&agg6[dst * 6 + k], x[src * 6 + k]);
}

// scatter of 128-wide features: agg[dst][:] += h1[src][:]
// thread t -> edge = t/32, 4 consecutive floats = (t%32)*4
__global__ void scatter128_kernel(const float* __restrict__ h1, const int* __restrict__ ei,
                                  int E, float* __restrict__ agg) {
    long long t = (long long)blockIdx.x * blockDim.x + threadIdx.x;
    if (t >= (long long)E * 32) return;
    int e  = (int)(t >> 5);
    int c4 = ((int)t & 31) * 4;
    int src = ei[e];
    int dst = ei[E + e];
    const float* hs = h1 + src * H + c4;
    float* ad = agg + dst * H + c4;
#pragma unroll
    for (int k = 0; k < 4; ++k) atomicAdd(ad + k, hs[k]);
}

// sparse interpolation: acc[rows[nz]][c] += vals[nz] * e[cols[nz]][c], c in [0,96)
// one block (96 threads) per nonzero
__global__ void interp_kernel(const float* __restrict__ vals, const int* __restrict__ cols,
                              const int* __restrict__ rows, const float* __restrict__ e,
                              float* __restrict__ acc) {
    int nz = blockIdx.x;
    int c  = threadIdx.x;
    float v = vals[nz];
    int col = cols[nz];
    int row = rows[nz];
    atomicAdd(&acc[row * F3 + c], v * e[col * F3 + c]);
}

// final decoder dot: out[r*3+branch] = bout + sum_c hdec[r][c]*Wout[c]
__global__ void dec_out_kernel(const float* __restrict__ hdec, const float* __restrict__ Wout,
                               const float* __restrict__ bout, float* __restrict__ out,
                               int n, int branch) {
    int r = blockIdx.x * blockDim.x + threadIdx.x;
    if (r >= n) return;
    float s = bout[0];
    const float* hr = hdec + r * DECW;
#pragma unroll
    for (int c = 0; c < DECW; ++c) s += hr[c] * Wout[c];
    out[r * 3 + branch] = s;
}

// ---------------------------------------------------------------------------
// WMMA fp32 GEMM (V_WMMA_F32_16X16X4_F32).
// One wave computes one 16x16 tile of D. A: (n,K) row-major, W: (K,Nc) row-major.
// A fragment (16x4 f32, ISA layout): lane l<16 holds M=l, K=k..k+1; lane>=16 holds
// M=l-16, K=k+2..k+3 -> per-lane v2f at A + row*K + half*2 + k (8B aligned).
// B fragment (4x16): VGPR0 lanes0-15 = row k, lanes16-31 = row k+2; VGPR1 = k+1/k+3
//   -> per-lane base W + col0 + m + half*2*Nc, elements at +k*Nc and +(k+1)*Nc.
// C/D: VGPR j, lanes0-15 -> M=j, lanes16-31 -> M=j+8, N=lane&15.
// Row OOB is handled by clamping to n-1 (unconditional loads; garbage rows are
// never stored), so the inner loop has NO exec-mask manipulation.
// K, Nc compile-time => fully unrolled, loads with immediate offsets.
// ---------------------------------------------------------------------------
template <int K, int Nc, int ACT, bool DUAL>
__global__ void gemm_wmma_kernel(const float* __restrict__ A0, const float* __restrict__ W0,
                                 const float* __restrict__ A1, const float* __restrict__ W1,
                                 const float* __restrict__ bias, float* __restrict__ D,
                                 int n) {
    const int lane = threadIdx.x & 31;
    const int wave = threadIdx.x >> 5;
    const int row0 = (blockIdx.x * 4 + wave) * 16;
    const int col0 = blockIdx.y * 16;
    if (row0 >= n) return;  // wave-uniform

    const int m    = lane & 15;
    const int half = lane >> 4;
    int row = row0 + m;
    const int rowc = (row < n) ? row : (n - 1);   // clamp: no divergent loads

    const float* a0p = A0 + rowc * K + half * 2;
    const float* w0p = W0 + col0 + m + half * 2 * Nc;
    const float* a1p = DUAL ? (A1 + rowc * K + half * 2) : nullptr;
    const float* w1p = DUAL ? (W1 + col0 + m + half * 2 * Nc) : nullptr;

    const float bv = bias[col0 + m];
    v8f acc = {bv, bv, bv, bv, bv, bv, bv, bv};

#pragma unroll
    for (int k = 0; k < K; k += 4) {
        v2f a = *(const v2f*)(a0p + k);
        v2f b;
        b.x = w0p[k * Nc];
        b.y = w0p[k * Nc + Nc];
        acc = __builtin_amdgcn_wmma_f32_16x16x4_f32(false, a, false, b, (short)0, acc,
                                                    false, false);
        if (DUAL) {
            v2f a2 = *(const v2f*)(a1p + k);
            v2f b2;
            b2.x = w1p[k * Nc];
            b2.y = w1p[k * Nc + Nc];
            acc = __builtin_amdgcn_wmma_f32_16x16x4_f32(false, a2, false, b2, (short)0, acc,
                                                        false, false);
        }
    }

#pragma unroll
    for (int j = 0; j < 8; ++j) {
        int r = row0 + j + half * 8;
        if (r < n) {
            float v = acc[j];
            if (ACT == 1) v = fmaxf(v, 0.0f);
            if (ACT == 2) v = (v > 0.0f) ? v : (expf(v) - 1.0f);
            D[r * Nc + col0 + m] = v;
        }
    }
}

// Decoder GEMM with implicit A = concat(e0[:,b*32:+32], i1[:,b*32:+32], i2[:,b*32:+32]).
// K=96, Nc=64, ELU. The 3-way source select depends only on k (each 4-wide K window
// stays inside one 32-block), so with full unroll it constant-folds to one of three
// precomputed base pointers.
__global__ void dec_gemm_wmma_kernel(const float* __restrict__ e0, const float* __restrict__ i1,
                                     const float* __restrict__ i2, const float* __restrict__ W,
                                     const float* __restrict__ bias, float* __restrict__ D,
                                     int n, int branch) {
    constexpr int K = F3, Nc = DECW;
    const int lane = threadIdx.x & 31;
    const int wave = threadIdx.x >> 5;
    const int row0 = (blockIdx.x * 4 + wave) * 16;
    const int col0 = blockIdx.y * 16;
    if (row0 >= n) return;

    const int m    = lane & 15;
    const int half = lane >> 4;
    int row = row0 + m;
    const int rowc = (row < n) ? row : (n - 1);

    const int abase = rowc * F3 + branch * 32 + half * 2;
    const float* s0 = e0 + abase;
    const float* s1 = i1 + abase;
    const float* s2 = i2 + abase;
    const float* wp = W + col0 + m + half * 2 * Nc;

    const float bv = bias[col0 + m];
    v8f acc = {bv, bv, bv, bv, bv, bv, bv, bv};

#pragma unroll
    for (int k = 0; k < K; k += 4) {
        const int blk = k >> 5;                       // compile-time after unroll
        const float* sp = (blk == 0) ? s0 : (blk == 1) ? s1 : s2;
        v2f a = *(const v2f*)(sp + (k & 31));
        v2f b;
        b.x = wp[k * Nc];
        b.y = wp[k * Nc + Nc];
        acc = __builtin_amdgcn_wmma_f32_16x16x4_f32(false, a, false, b, (short)0, acc,
                                                    false, false);
    }

#pragma unroll
    for (int j = 0; j < 8; ++j) {
        int r = row0 + j + half * 8;
        if (r < n) {
            float v = acc[j];
            v = (v > 0.0f) ? v : (expf(v) - 1.0f);  // ELU
            D[r * Nc + col0 + m] = v;
        }
    }
}

// ---------------------------------------------------------------------------
// Host-side orchestration
// ---------------------------------------------------------------------------
static inline dim3 gemm_grid(int n, int Nc) {
    int row_tiles = (n + 15) / 16;
    return dim3((row_tiles + 3) / 4, Nc / 16, 1);
}

extern "C" void kernel_launch(void* const* d_in, const int* in_sizes, int n_in,
                              void* d_out, int out_size, void* d_ws, size_t ws_size,
                              hipStream_t stream) {
    (void)in_sizes; (void)n_in; (void)out_size; (void)ws_size;

    // ------- inputs (setup_inputs order) -------
    const float* x[3]   = {(const float*)d_in[0], (const float*)d_in[1], (const float*)d_in[2]};
    const int*   ei[3]  = {(const int*)d_in[3], (const int*)d_in[4], (const int*)d_in[5]};
    const int*   A1r    = (const int*)d_in[6];
    const int*   A1c    = (const int*)d_in[7];
    const float* A1v    = (const float*)d_in[8];
    const int*   A2r    = (const int*)d_in[9];
    const int*   A2c    = (const int*)d_in[10];
    const float* A2v    = (const float*)d_in[11];
    const float* c1Wrel  = (const float*)d_in[12];   // (3,6,128)
    const float* c1brel  = (const float*)d_in[13];   // (3,128)
    const float* c1Wroot = (const float*)d_in[14];   // (3,6,128)
    const float* c2Wrel  = (const float*)d_in[15];   // (3,128,128)
    const float* c2brel  = (const float*)d_in[16];   // (3,128)
    const float* c2Wroot = (const float*)d_in[17];   // (3,128,128)
    const float* linW    = (const float*)d_in[18];   // (3,128,96)
    const float* linb    = (const float*)d_in[19];   // (3,96)
    const float* decW0   = (const float*)d_in[20];   // (3,96,64)
    const float* decb0   = (const float*)d_in[21];   // (3,64)
    const float* decWout = (const float*)d_in[22];   // (3,64,1)
    const float* decbout = (const float*)d_in[23];   // (3,1)
    float* out = (float*)d_out;                      // (N0,3,1)

    // ------- workspace layout (floats) -------
    float* ws = (float*)d_ws;
    float* h1   = ws;                          // N0*128 (reused across levels)
    float* agg  = h1   + (size_t)N0 * H;       // N0*128
    float* h2   = agg  + (size_t)N0 * H;       // N0*128
    float* agg6 = h2   + (size_t)N0 * H;       // N0*6
    float* e0   = agg6 + (size_t)N0 * 6;       // N0*96
    float* e1   = e0   + (size_t)N0 * F3;      // N1*96
    float* e2   = e1   + (size_t)N1 * F3;      // N2*96
    float* i1   = e2   + (size_t)N2 * F3;      // N0*96
    float* i2   = i1   + (size_t)N0 * F3;      // N0*96
    float* hdec = i2   + (size_t)N0 * F3;      // N0*64

    const int nlv[3] = {N0, N1, N2};
    const int Elv[3] = {E0, E1, E2};
    float* elv[3] = {e0, e1, e2};

    // ------- encoders -------
    for (int i = 0; i < 3; ++i) {
        const int n = nlv[i];
        const int E = Elv[i];

        // conv1: agg6 = segment_sum(x[src], dst); h1 = relu(agg6@Wrel + brel + x@Wroot)
        {
            long long cnt = (long long)n * 6;
            zero_kernel<<<(int)((cnt + 255) / 256), 256, 0, stream>>>(agg6, cnt);
            scatter6_kernel<<<(E + 255) / 256, 256, 0, stream>>>(x[i], ei[i], E, agg6);
            conv1_kernel<<<n, H, 0, stream>>>(x[i], agg6,
                                              c1Wrel + (size_t)i * 6 * H,
                                              c1brel + (size_t)i * H,
                                              c1Wroot + (size_t)i * 6 * H, h1, n);
        }

        // conv2: agg = segment_sum(h1[src], dst); h2 = relu(agg@Wrel2 + h1@Wroot2 + brel2)
        {
            long long cnt = (long long)n * H;
            zero_kernel<<<(int)((cnt + 255) / 256), 256, 0, stream>>>(agg, cnt);
            long long th = (long long)E * 32;
            scatter128_kernel<<<(int)((th + 255) / 256), 256, 0, stream>>>(h1, ei[i], E, agg);
            gemm_wmma_kernel<H, H, 1, true><<<gemm_grid(n, H), 128, 0, stream>>>(
                agg, c2Wrel + (size_t)i * H * H,
                h1,  c2Wroot + (size_t)i * H * H,
                c2brel + (size_t)i * H, h2, n);
        }

        // lin: e_i = h2 @ linW + linb
        gemm_wmma_kernel<H, F3, 0, false><<<gemm_grid(n, F3), 128, 0, stream>>>(
            h2, linW + (size_t)i * H * F3, nullptr, nullptr,
            linb + (size_t)i * F3, elv[i], n);
    }

    // ------- sparse interpolations into level-0 rows -------
    {
        long long cnt = (long long)N0 * F3;
        zero_kernel<<<(int)((cnt + 255) / 256), 256, 0, stream>>>(i1, cnt);
        zero_kernel<<<(int)((cnt + 255) / 256), 256, 0, stream>>>(i2, cnt);
        interp_kernel<<<NNZ, F3, 0, stream>>>(A1v, A1c, A1r, e1, i1);
        interp_kernel<<<NNZ, F3, 0, stream>>>(A2v, A2c, A2r, e2, i2);
    }

    // ------- decoders (interped A matrix read implicitly from e0/i1/i2) -------
    for (int b = 0; b < 3; ++b) {
        dec_gemm_wmma_kernel<<<gemm_grid(N0, DECW), 128, 0, stream>>>(
            e0, i1, i2, decW0 + (size_t)b * F3 * DECW, decb0 + (size_t)b * DECW,
            hdec, N0, b);
        dec_out_kernel<<<(N0 + 255) / 256, 256, 0, stream>>>(
            hdec, decWout + (size_t)b * DECW, decbout + b, out, N0, b);
    }
}